// EncoderVAE_11012296147415
// MI455X (gfx1250) — compile-verified
//
#include <hip/hip_runtime.h>
#include <math.h>

typedef float v2f __attribute__((ext_vector_type(2)));
typedef float v8f __attribute__((ext_vector_type(8)));
typedef int   v4i __attribute__((vector_size(16)));   // matches async-LDS builtin param

#define N_NODES 50000
#define N_EDGES 800000
#define ROW_TILES 3125   /* 50000 / 16 exactly */

#define AS1 __attribute__((address_space(1)))
#define AS3 __attribute__((address_space(3)))

#if __has_builtin(__builtin_amdgcn_global_load_async_to_lds_b128) && \
    __has_builtin(__builtin_amdgcn_s_wait_asynccnt)
#define HAVE_ASYNC_LDS 1
#else
#define HAVE_ASYNC_LDS 0
#endif

// ---------------------------------------------------------------- utilities
__global__ void fill_kernel(float* __restrict__ p, float v, int n) {
  int i = blockIdx.x * blockDim.x + threadIdx.x;
  if (i < n) p[i] = v;
}

__global__ void degree_kernel(const int* __restrict__ dst, float* __restrict__ deg, int e) {
  int i = blockIdx.x * blockDim.x + threadIdx.x;
  if (i < e) atomicAdd(&deg[dst[i]], 1.0f);
}

// ---------------------------------------------------------------- WMMA GEMM
// H[rows,NCOLS] = X[rows,128] @ W[128,NCOLS], fp32-exact via V_WMMA_F32_16X16X4_F32.
// W is staged TRANSPOSED in LDS with an XOR swizzle so each B fragment is one
// aligned ds_load_b64 into an even VGPR pair (no shuffle movs, no bank storm).
template <int NCOLS>
__global__ __launch_bounds__(128) void gemm_wmma_kernel(const float* __restrict__ X,
                                                        const float* __restrict__ W,
                                                        float* __restrict__ H,
                                                        int rowTiles) {
  constexpr int CT = NCOLS / 16;
  __shared__ float Ws[NCOLS * 128];  // Ws[col][k], k swizzled
  for (int i = threadIdx.x; i < 128 * NCOLS; i += 128) {
    int k = i / NCOLS;   // row of W (K index)
    int n = i % NCOLS;   // col of W (N index)
    Ws[n * 128 + (k ^ ((n & 31) << 1))] = W[i];
  }
  __syncthreads();

  const int wave = threadIdx.x >> 5;
  const int lane = threadIdx.x & 31;
  const int tile = blockIdx.x * 4 + wave;
  if (tile >= rowTiles) return;          // whole-wave exit; EXEC stays all-ones

  const int laneM    = lane & 15;
  const int laneHalf = lane >> 4;        // 0: K pair {0,1}, 1: K pair {2,3}
  const int rowBase  = tile * 16;
  const float* Xrow  = X + (size_t)(rowBase + laneM) * 128;

  v8f acc[CT];
#pragma unroll
  for (int c = 0; c < CT; ++c) acc[c] = {};

  for (int k = 0; k < 32; ++k) {
    const int k0 = k * 4 + laneHalf * 2;             // even
    v2f a = *(const v2f*)(Xrow + k0);                // aligned global b64
#pragma unroll
    for (int c = 0; c < CT; ++c) {
      const int col = c * 16 + laneM;
      v2f b = *(const v2f*)&Ws[col * 128 + (k0 ^ ((col & 31) << 1))];  // aligned ds b64
      acc[c] = __builtin_amdgcn_wmma_f32_16x16x4_f32(
          false, a, false, b, (short)0, acc[c], false, false);
    }
  }

  // C/D layout: vgpr v, lanes 0-15 -> M=v, lanes 16-31 -> M=v+8; N = laneM
#pragma unroll
  for (int c = 0; c < CT; ++c)
#pragma unroll
    for (int v = 0; v < 8; ++v)
      H[(size_t)(rowBase + v + 8 * laneHalf) * NCOLS + c * 16 + laneM] = acc[c][v];
}

// ---------------------------------------------------------------- edge scatter
template <int F>
__global__ void scatter_kernel(const int* __restrict__ src, const int* __restrict__ dst,
                               const float* __restrict__ deg, const float* __restrict__ h,
                               float* __restrict__ agg, int e) {
  constexpr int QF = F / 4;
  int idx  = blockIdx.x * blockDim.x + threadIdx.x;
  int eIdx = idx / QF;
  if (eIdx >= e) return;
  int f = (idx % QF) * 4;
  // prefetch the gather row a few thousand edges ahead (global_prefetch_b8)
  int ePf = eIdx + 4096;
  if (ePf < e) __builtin_prefetch(h + (size_t)src[ePf] * F + f, 0, 1);
  int s = src[eIdx], d = dst[eIdx];
  float norm = rsqrtf(deg[s]) * rsqrtf(deg[d]);
  const float4 hv = *(const float4*)(h + (size_t)s * F + f);
  float* ap = agg + (size_t)d * F + f;
  atomicAdd(ap + 0, norm * hv.x);
  atomicAdd(ap + 1, norm * hv.y);
  atomicAdd(ap + 2, norm * hv.z);
  atomicAdd(ap + 3, norm * hv.w);
}

// h[i,f] = tanh(agg[i,f] + h[i,f]/deg[i] + bias[f])   (in place on h)
template <int F>
__global__ void combine_kernel(const float* __restrict__ agg, float* __restrict__ h,
                               const float* __restrict__ deg, const float* __restrict__ bias,
                               int total) {
  int idx = blockIdx.x * blockDim.x + threadIdx.x;
  if (idx >= total) return;
  int row = idx / F;
  int f   = idx % F;
  float v = agg[idx] + h[idx] / deg[row] + bias[f];
  h[idx] = tanhf(v);
}

// ---------------------------------------------------------------- product reduce
__global__ __launch_bounds__(256) void prod_partial_kernel(const float* __restrict__ h2,
                                                           float* __restrict__ part,
                                                           int nrows) {
  __shared__ float lds[256];
  int f   = threadIdx.x & 63;
  int grp = threadIdx.x >> 6;
  int chunk = (nrows + gridDim.x - 1) / gridDim.x;
  int start = blockIdx.x * chunk;
  int end   = start + chunk; if (end > nrows) end = nrows;
  float p = 1.0f;
  for (int i = start + grp; i < end; i += 4) p *= h2[(size_t)i * 64 + f];
  lds[threadIdx.x] = p;
  __syncthreads();
  if (grp == 0)
    part[blockIdx.x * 64 + f] = lds[f] * lds[64 + f] * lds[128 + f] * lds[192 + f];
}

__global__ void prod_final_kernel(const float* __restrict__ part, float* __restrict__ g,
                                  int nblocks) {
  int f = threadIdx.x;  // 64 threads
  float p = 1.0f;
  for (int b = 0; b < nblocks; ++b) p *= part[b * 64 + f];
  g[f] = p;
}

// ---------------------------------------------------------------- MLP head
// Wmu / Wlv columns are reused 32x -> stage them in LDS; the gfx1250 async
// global->LDS path (ASYNCcnt) issues the Wmu staging up front, overlapped
// with the d1/d2/gb layers.
__device__ __forceinline__ void stage_w_lds(const float* __restrict__ Wg,
                                            float* __restrict__ sW, int t) {
#if HAVE_ASYNC_LDS
  for (int i = t * 4; i < 128 * 64; i += 256 * 4)
    __builtin_amdgcn_global_load_async_to_lds_b128(
        (AS1 v4i*)(Wg + i), (AS3 v4i*)&sW[i], 0, 0);
#else
  for (int i = t; i < 128 * 64; i += 256) sW[i] = Wg[i];
#endif
}

__device__ __forceinline__ void stage_w_wait() {
#if HAVE_ASYNC_LDS
  __builtin_amdgcn_s_wait_asynccnt(0);
#endif
}

__global__ __launch_bounds__(256) void head_kernel(
    const float* __restrict__ g_in,
    const float* __restrict__ Wd1, const float* __restrict__ bd1,
    const float* __restrict__ Wd2, const float* __restrict__ bd2,
    const float* __restrict__ Wb,  const float* __restrict__ bb,
    const float* __restrict__ Wmu, const float* __restrict__ bmu,
    const float* __restrict__ Wlv, const float* __restrict__ blv,
    const float* __restrict__ eps, float* __restrict__ out) {
  __shared__ float sg[64], sg1[256], sg2[128], sgb[4096];
  __shared__ float sWm[128 * 64];  // holds Wmu, then Wlv (two phases)
  const int t = threadIdx.x;

  stage_w_lds(Wmu, sWm, t);        // async: overlap with layers below
  if (t < 64) sg[t] = g_in[t];
  __syncthreads();
  {  // g1 = tanh(g @ Wd1 + bd1), 256 outputs
    float acc = bd1[t];
    for (int k = 0; k < 64; ++k) acc += sg[k] * Wd1[k * 256 + t];
    sg1[t] = tanhf(acc);
  }
  __syncthreads();
  if (t < 128) {  // g2 = tanh(g1 @ Wd2 + bd2), 128 outputs
    float acc = bd2[t];
    for (int k = 0; k < 256; ++k) acc += sg1[k] * Wd2[k * 128 + t];
    sg2[t] = tanhf(acc);
  }
  __syncthreads();
  for (int j = t; j < 4096; j += 256) {  // gb = g2 @ Wb + bb  -> [32,128]
    float acc = bb[j];
    for (int k = 0; k < 128; ++k) acc += sg2[k] * Wb[k * 4096 + j];
    sgb[j] = acc;
  }
  stage_w_wait();                  // Wmu staged by now
  __syncthreads();

  float am[8];
#pragma unroll
  for (int j = 0; j < 8; ++j) {    // mu = gb @ Wmu + bmu
    int o = t + j * 256;
    int b = o >> 6, f = o & 63;
    float acc = bmu[f];
    const float* gb = &sgb[b * 128];
    for (int k = 0; k < 128; ++k) acc += gb[k] * sWm[k * 64 + f];
    am[j] = acc;
  }
  __syncthreads();                 // everyone done reading sWm(Wmu)
  stage_w_lds(Wlv, sWm, t);
  stage_w_wait();
  __syncthreads();

#pragma unroll
  for (int j = 0; j < 8; ++j) {    // lv = gb @ Wlv + blv ; z = eps*exp(lv/2)+mu
    int o = t + j * 256;
    int b = o >> 6, f = o & 63;
    float acc = blv[f];
    const float* gb = &sgb[b * 128];
    for (int k = 0; k < 128; ++k) acc += gb[k] * sWm[k * 64 + f];
    float z = eps[o] * __expf(0.5f * acc) + am[j];
    out[o]        = z;
    out[2048 + o] = am[j];
    out[4096 + o] = acc;
  }
}

// ---------------------------------------------------------------- launch
extern "C" void kernel_launch(void* const* d_in, const int* in_sizes, int n_in,
                              void* d_out, int out_size, void* d_ws, size_t ws_size,
                              hipStream_t stream) {
  const float* x   = (const float*)d_in[0];
  const int*   ei  = (const int*)  d_in[1];
  const float* eps = (const float*)d_in[2];
  const float* W1  = (const float*)d_in[3];
  const float* b1  = (const float*)d_in[4];
  const float* W2  = (const float*)d_in[5];
  const float* b2  = (const float*)d_in[6];
  const float* Wd1 = (const float*)d_in[7];
  const float* bd1 = (const float*)d_in[8];
  const float* Wd2 = (const float*)d_in[9];
  const float* bd2 = (const float*)d_in[10];
  const float* Wb  = (const float*)d_in[11];
  const float* bb  = (const float*)d_in[12];
  const float* Wmu = (const float*)d_in[13];
  const float* bmu = (const float*)d_in[14];
  const float* Wlv = (const float*)d_in[15];
  const float* blv = (const float*)d_in[16];
  float* out = (float*)d_out;

  const int* src = ei;
  const int* dst = ei + N_EDGES;

  // workspace layout (floats); agg1 region reused for h2+agg2 after combine1
  float* ws   = (float*)d_ws;
  float* deg  = ws;                                 // 50048 (padded)
  float* h1   = ws + 50048;                         // N*128
  float* agg1 = h1 + (size_t)N_NODES * 128;         // N*128
  float* h2   = agg1;                               // N*64  (reuse)
  float* agg2 = agg1 + (size_t)N_NODES * 64;        // N*64  (reuse)
  float* part = agg1 + (size_t)N_NODES * 128;       // 128*64
  float* gbuf = part + 128 * 64;                    // 64

  const int TPB = 256;

  // deg = 1 (self loop), agg1 = 0
  fill_kernel<<<(N_NODES + TPB - 1) / TPB, TPB, 0, stream>>>(deg, 1.0f, N_NODES);
  fill_kernel<<<(N_NODES * 128 + TPB - 1) / TPB, TPB, 0, stream>>>(agg1, 0.0f, N_NODES * 128);
  degree_kernel<<<(N_EDGES + TPB - 1) / TPB, TPB, 0, stream>>>(dst, deg, N_EDGES);

  // ---- layer 1: h1 = X @ W1 ; scatter ; combine+tanh
  gemm_wmma_kernel<128><<<(ROW_TILES + 3) / 4, 128, 0, stream>>>(x, W1, h1, ROW_TILES);
  {
    int total = N_EDGES * (128 / 4);
    scatter_kernel<128><<<(total + TPB - 1) / TPB, TPB, 0, stream>>>(src, dst, deg, h1, agg1, N_EDGES);
  }
  combine_kernel<128><<<(N_NODES * 128 + TPB - 1) / TPB, TPB, 0, stream>>>(agg1, h1, deg, b1, N_NODES * 128);

  // ---- layer 2: h2 = h1 @ W2 ; scatter ; combine+tanh
  fill_kernel<<<(N_NODES * 64 + TPB - 1) / TPB, TPB, 0, stream>>>(agg2, 0.0f, N_NODES * 64);
  gemm_wmma_kernel<64><<<(ROW_TILES + 3) / 4, 128, 0, stream>>>(h1, W2, h2, ROW_TILES);
  {
    int total = N_EDGES * (64 / 4);
    scatter_kernel<64><<<(total + TPB - 1) / TPB, TPB, 0, stream>>>(src, dst, deg, h2, agg2, N_EDGES);
  }
  combine_kernel<64><<<(N_NODES * 64 + TPB - 1) / TPB, TPB, 0, stream>>>(agg2, h2, deg, b2, N_NODES * 64);

  // ---- node product -> g[64]
  prod_partial_kernel<<<128, 256, 0, stream>>>(h2, part, N_NODES);
  prod_final_kernel<<<1, 64, 0, stream>>>(part, gbuf, 128);

  // ---- MLP head -> z, mu, lv
  head_kernel<<<1, 256, 0, stream>>>(gbuf, Wd1, bd1, Wd2, bd2, Wb, bb,
                                     Wmu, bmu, Wlv, blv, eps, out);
}